// MaximumMeanDiscrepancy_9646496547399
// MI455X (gfx1250) — compile-verified
//
#include <hip/hip_runtime.h>
#include <hip/hip_bf16.h>

typedef __attribute__((ext_vector_type(2))) float v2f;
typedef __attribute__((ext_vector_type(8))) float v8f;

#define BATCHN 160
#define FEATN  1024
#define NPAIRS 1522
#define BC_ROWS (NPAIRS * 4)   /* 6088 */
#define WC_ROWS 160

// coef[k] = -log2(e) / (2 * sigma_k)
static __device__ const float kCoef[19] = {
  -7.2134752e+05f, -7.2134752e+04f, -7.2134752e+03f, -7.2134752e+02f,
  -7.2134752e+01f, -7.2134752e+00f, -7.2134752e-01f, -1.4426950e-01f,
  -7.2134752e-02f, -4.8089835e-02f, -3.6067376e-02f, -2.8853901e-02f,
  -2.4044917e-02f, -2.0612786e-02f, -7.2134752e-03f, -7.2134752e-04f,
  -7.2134752e-05f, -7.2134752e-06f, -7.2134752e-07f
};

__global__ void mmd_init_out(float* out) {
  if (threadIdx.x < 2) out[threadIdx.x] = 0.0f;
}

// One wave (32 lanes) per row; grid = (160, 2) for source/target.
__global__ void mmd_row_norms(const float* __restrict__ S, const float* __restrict__ T,
                              float* __restrict__ ns, float* __restrict__ nt) {
  const float* F = blockIdx.y ? T : S;
  float*       n = blockIdx.y ? nt : ns;
  const int row = blockIdx.x;
  const float* p = F + (size_t)row * FEATN;
  float s = 0.0f;
  for (int j = threadIdx.x; j < FEATN; j += 32) { float v = p[j]; s += v * v; }
  for (int m = 16; m; m >>= 1) s += __shfl_xor(s, m, 32);
  if (threadIdx.x == 0) n[row] = s;
}

// G = F * F^T via V_WMMA_F32_16X16X4_F32, K-loop over 1024.
// Grid = (10, 10, 2); one wave per 16x16 tile.
__global__ void mmd_gram_wmma(const float* __restrict__ S, const float* __restrict__ T,
                              float* __restrict__ Gs, float* __restrict__ Gt) {
  const float* F = blockIdx.z ? T : S;
  float*       G = blockIdx.z ? Gt : Gs;
  const int lane = threadIdx.x;
  const int m  = lane & 15;          // A: row M   / B: col N
  const int hi = lane >> 4;          // selects K pair {0,1} vs {2,3}
  const int rowA = blockIdx.y * 16 + m;
  const int rowB = blockIdx.x * 16 + m;
  const v2f* A2 = (const v2f*)(F + (size_t)rowA * FEATN);
  const v2f* B2 = (const v2f*)(F + (size_t)rowB * FEATN);
  v8f c = {};
#pragma unroll 4
  for (int k2 = 0; k2 < FEATN / 2; k2 += 2) {   // 256 WMMA steps of K=4
    v2f a = A2[k2 + hi];
    v2f b = B2[k2 + hi];
    c = __builtin_amdgcn_wmma_f32_16x16x4_f32(false, a, false, b, (short)0, c,
                                              false, false);
  }
  const int rowBase = blockIdx.y * 16;
  const int colBase = blockIdx.x * 16;
#pragma unroll
  for (int r = 0; r < 8; ++r) {
    const int M = r + 8 * hi;                       // C layout: M = r + 8*(lane>=16)
    G[(size_t)(rowBase + M) * BATCHN + colBase + m] = c[r];   // N = lane&15
  }
}

// Build wc [160x4], bc [6088x4] plus per-row squared norms, for both matrices.
__global__ void mmd_gather_blocks(const float* __restrict__ Gs, const float* __restrict__ Gt,
                                  const float* __restrict__ ns, const float* __restrict__ nt,
                                  float* wcs, float* wct, float* wcsn, float* wctn,
                                  float* bcs, float* bct, float* bcsn, float* bctn) {
  const int tid = blockIdx.x * blockDim.x + threadIdx.x;
  const int per = WC_ROWS + BC_ROWS;
  if (tid >= 2 * per) return;
  const int mat = tid / per;
  const int r   = tid % per;
  const float* G  = mat ? Gt : Gs;
  const float* n  = mat ? nt : ns;
  float* wc  = mat ? wct  : wcs;
  float* wcn = mat ? wctn : wcsn;
  float* bc  = mat ? bct  : bcs;
  float* bcn = mat ? bctn : bcsn;
  if (r < WC_ROWS) {
    const int p = r >> 2;           // r == p*4 + i
    const float nr = n[r];
    float norm = 0.0f;
#pragma unroll
    for (int j = 0; j < 4; ++j) {
      const int gc = p * 4 + j;
      const float v = nr + n[gc] - 2.0f * G[(size_t)r * BATCHN + gc];
      wc[r * 4 + j] = v;
      norm += v * v;
    }
    wcn[r] = norm;
  } else {
    const int rb = r - WC_ROWS;
    const int k = rb >> 2, i = rb & 3;
    int pi, pj;
    if (k == 0) { pi = 0; pj = 1; }
    else {
      const int t = k - 1;
      pi = 1 + t / 39;
      const int jj = t % 39;
      pj = jj + (jj >= pi ? 1 : 0);
    }
    const int gr = pi * 4 + i;
    const float nr = n[gr];
    float norm = 0.0f;
#pragma unroll
    for (int j = 0; j < 4; ++j) {
      const int gc = pj * 4 + j;
      const float v = nr + n[gc] - 2.0f * G[(size_t)gr * BATCHN + gc];
      bc[rb * 4 + j] = v;
      norm += v * v;
    }
    bcn[rb] = norm;
  }
}

// sum_{pairs} sum_k exp(-d/(2 s_k)) over rows of X [N x 4] vs rows of Y [N x 4].
// Cross products for a 16x16 tile come from ONE V_WMMA_F32_16X16X4_F32 (K=4).
// One wave per tile; partial (scaled) sums accumulate into *out via atomicAdd.
__global__ void mmd_gk_tile_wmma(const float* __restrict__ X, const float* __restrict__ Xn,
                                 const float* __restrict__ Y, const float* __restrict__ Yn,
                                 int N, float scale, float* __restrict__ out) {
  const int lane = threadIdx.x;
  const int m  = lane & 15;
  const int hi = lane >> 4;
  const int rowA0 = blockIdx.y * 16;
  const int rowB0 = blockIdx.x * 16;
  const int ra = rowA0 + m;
  const int rb = rowB0 + m;
  v2f a = {0.0f, 0.0f}, b = {0.0f, 0.0f};
  if (ra < N) a = ((const v2f*)X)[ra * 2 + hi];   // A 16x4: K pair per half-wave
  if (rb < N) b = ((const v2f*)Y)[rb * 2 + hi];   // B  4x16: same striping
  v8f c = {};
  c = __builtin_amdgcn_wmma_f32_16x16x4_f32(false, a, false, b, (short)0, c,
                                            false, false);
  const bool colValid = rb < N;
  const float nb = colValid ? Yn[rb] : 0.0f;      // column N = lane&15
  float acc = 0.0f;
#pragma unroll
  for (int r = 0; r < 8; ++r) {
    const int M = r + 8 * hi;
    const int rar = rowA0 + M;
    const float na = (rar < N) ? Xn[rar] : 0.0f;
    const float d = na + nb - 2.0f * c[r];
    float s = 0.0f;
#pragma unroll
    for (int k = 0; k < 19; ++k) s += __builtin_amdgcn_exp2f(d * kCoef[k]);
    if ((rar < N) && colValid) acc += s;
  }
  for (int msk = 16; msk; msk >>= 1) acc += __shfl_xor(acc, msk, 32);
  if (lane == 0) atomicAdd(out, acc * scale);
}

extern "C" void kernel_launch(void* const* d_in, const int* in_sizes, int n_in,
                              void* d_out, int out_size, void* d_ws, size_t ws_size,
                              hipStream_t stream) {
  (void)in_sizes; (void)n_in; (void)out_size; (void)ws_size;
  const float* S = (const float*)d_in[0];   // source_features [160,1024]
  const float* T = (const float*)d_in[1];   // target_features [160,1024]
  float* out = (float*)d_out;               // out[0]=wc mmd, out[1]=bc mmd
  float* ws  = (float*)d_ws;

  // Workspace layout (floats): ~114k floats (~456 KB)
  float* ns   = ws;                 // 160
  float* nt   = ns   + BATCHN;      // 160
  float* Gs   = nt   + BATCHN;      // 25600
  float* Gt   = Gs   + BATCHN * BATCHN;
  float* wcs  = Gt   + BATCHN * BATCHN;   // 640
  float* wct  = wcs  + WC_ROWS * 4;
  float* wcsn = wct  + WC_ROWS * 4;       // 160
  float* wctn = wcsn + WC_ROWS;
  float* bcs  = wctn + WC_ROWS;           // 24352
  float* bct  = bcs  + BC_ROWS * 4;
  float* bcsn = bct  + BC_ROWS * 4;       // 6088
  float* bctn = bcsn + BC_ROWS;

  mmd_init_out<<<1, 32, 0, stream>>>(out);
  mmd_row_norms<<<dim3(BATCHN, 2), 32, 0, stream>>>(S, T, ns, nt);
  mmd_gram_wmma<<<dim3(BATCHN / 16, BATCHN / 16, 2), 32, 0, stream>>>(S, T, Gs, Gt);
  const int totalRows = 2 * (WC_ROWS + BC_ROWS);
  mmd_gather_blocks<<<(totalRows + 255) / 256, 256, 0, stream>>>(
      Gs, Gt, ns, nt, wcs, wct, wcsn, wctn, bcs, bct, bcsn, bctn);

  // wc mmd -> out[0]
  const float wsc = 1.0f / ((float)WC_ROWS * (float)WC_ROWS);
  dim3 gwc(WC_ROWS / 16, WC_ROWS / 16);   // 10x10
  mmd_gk_tile_wmma<<<gwc, 32, 0, stream>>>(wcs, wcsn, wcs, wcsn, WC_ROWS,  wsc,        out + 0);
  mmd_gk_tile_wmma<<<gwc, 32, 0, stream>>>(wct, wctn, wct, wctn, WC_ROWS,  wsc,        out + 0);
  mmd_gk_tile_wmma<<<gwc, 32, 0, stream>>>(wcs, wcsn, wct, wctn, WC_ROWS, -2.0f * wsc, out + 0);

  // bc mmd -> out[1]
  const float bsc = 1.0f / ((float)BC_ROWS * (float)BC_ROWS);
  const int bt = (BC_ROWS + 15) / 16;     // 381
  dim3 gbc(bt, bt);
  mmd_gk_tile_wmma<<<gbc, 32, 0, stream>>>(bcs, bcsn, bcs, bcsn, BC_ROWS,  bsc,        out + 1);
  mmd_gk_tile_wmma<<<gbc, 32, 0, stream>>>(bct, bctn, bct, bctn, BC_ROWS,  bsc,        out + 1);
  mmd_gk_tile_wmma<<<gbc, 32, 0, stream>>>(bcs, bcsn, bct, bctn, BC_ROWS, -2.0f * bsc, out + 1);
}